// LSTM_24610162606633
// MI455X (gfx1250) — compile-verified
//
#include <hip/hip_runtime.h>
#include <hip/hip_bf16.h>

// LSTM: B=4096, T=2048, H=2, in=1.  One wave handles a 16-row batch tile.
// Transposed WMMA formulation (v_wmma_f32_16x16x4_f32):
//   gates^T(8x16) = W(16x4) @ [h0; h1; x; 1](4x16)
// A = constant weight matrix (rows = gates, rows 8-15 zero), with sigmoid-gate
// rows PRE-SCALED by 0.5 so sigmoid(z) = 0.5*tanh(z/2)+0.5 needs no mul.
// B = per-step state matrix (columns = batch rows).  D lands lane-major
// (VGPR v, lane m = gate v of row m): no transpose, no LDS on the chain.
// Activations use native v_tanh_f32 when the toolchain exposes it
// (__has_builtin-guarded; exact exp2/rcp fallback otherwise).
// x: register double-buffer across 32-step chunks + quad-pipelined LDS reads.
// Outputs staged in LDS, flushed fully coalesced.

typedef float v2f __attribute__((ext_vector_type(2)));
typedef float v8f __attribute__((ext_vector_type(8)));

#define B_SZ  4096
#define T_SZ  2048
#define TCH   32     // time chunk staged through LDS
#define XP4   36     // staging row stride (words): 16B-aligned rows
#define WAVES 4      // waves (tiles) per block

__device__ __forceinline__ float sw_tanh(float x) {
    // tanh(x) = 2/(1+exp2(-2*log2e*x)) - 1   (v_exp_f32 + v_rcp_f32)
    float e = __builtin_amdgcn_exp2f(-2.8853900817779268f * x);
    return __builtin_fmaf(2.0f, __builtin_amdgcn_rcpf(1.0f + e), -1.0f);
}

#if __has_builtin(__builtin_amdgcn_tanhf)
#define HW_TANH(x) __builtin_amdgcn_tanhf(x)
#elif __has_builtin(__builtin_amdgcn_tanh_f32)
#define HW_TANH(x) __builtin_amdgcn_tanh_f32(x)
#else
#define HW_TANH(x) sw_tanh(x)
#endif

// sigmoid on a pre-halved argument: sigmoid(2z) = 0.5*tanh(z)+0.5
__device__ __forceinline__ float sig_prescaled(float zhalf) {
    return __builtin_fmaf(0.5f, HW_TANH(zhalf), 0.5f);
}

// One LSTM timestep: WMMA produces lane-major gates in d_[0..7].
// Sigmoid rows of A are pre-scaled by 0.5; g rows (4,5) are full-scale.
#define STEP(XS, LOUT, POUT) do {                                             \
    v2f b_;                                                                   \
    b_.x = hi ? (XS) : h0;                                                    \
    b_.y = hi ? 1.0f : h1;                                                    \
    v8f d_ = __builtin_amdgcn_wmma_f32_16x16x4_f32(                           \
                 false, amat, false, b_, (short)0, cz, false, false);         \
    float i0_ = sig_prescaled(d_[0]), i1_ = sig_prescaled(d_[1]);             \
    float f0_ = sig_prescaled(d_[2]), f1_ = sig_prescaled(d_[3]);             \
    float g0_ = HW_TANH(d_[4]),       g1_ = HW_TANH(d_[5]);                   \
    float o0_ = sig_prescaled(d_[6]), o1_ = sig_prescaled(d_[7]);             \
    c0 = __builtin_fmaf(f0_, c0, i0_ * g0_);                                  \
    c1 = __builtin_fmaf(f1_, c1, i1_ * g1_);                                  \
    h0 = o0_ * HW_TANH(c0);                                                   \
    h1 = o1_ * HW_TANH(c1);                                                   \
    (LOUT) = __builtin_fmaf(wo1, h1, __builtin_fmaf(wo0, h0, bo));            \
    (POUT) = sig_prescaled(0.5f * (LOUT));                                    \
} while (0)

__global__ __launch_bounds__(32 * WAVES, 1)
void lstm_wmma_kernel(const float* __restrict__ x,
                      const float* __restrict__ W_ih,   // [8]
                      const float* __restrict__ W_hh,   // [8][2] row-major
                      const float* __restrict__ b_ih,   // [8]
                      const float* __restrict__ b_hh,   // [8]
                      const float* __restrict__ W_out,  // [2]
                      const float* __restrict__ b_out,  // [1]
                      float* __restrict__ out)          // logits | preds
{
    __shared__ float xbuf[WAVES][16 * XP4];  // staged inputs  [row][t]
    __shared__ float lbuf[WAVES][32 * XP4];  // staged logits (rows 16-31 junk)
    __shared__ float pbuf[WAVES][32 * XP4];  // staged preds  (rows 16-31 junk)

    const int lane = threadIdx.x & 31;
    const int w    = threadIdx.x >> 5;
    const int b0   = (blockIdx.x * WAVES + w) * 16;   // batch tile base
    const int n    = lane & 15;
    const bool hi  = lane >= 16;

    // ---- Constant A operand = weight matrix (16x4, row = gate, rows 8-15 0):
    //   VGPR0: lanes0-15 K=0 (Whh[g][0]), lanes16-31 K=2 (Wih[g])
    //   VGPR1: lanes0-15 K=1 (Whh[g][1]), lanes16-31 K=3 (bih[g]+bhh[g])
    // Sigmoid-gate rows (g = 0,1,2,3,6,7) pre-scaled by 0.5 (exact:
    // sigmoid(z) = 0.5*tanh(z/2)+0.5); tanh rows (g = 4,5) full scale.
    v2f amat;
    if (n < 8) {
        const float sc = (n == 4 || n == 5) ? 1.0f : 0.5f;
        amat.x = sc * (hi ? W_ih[n]             : W_hh[n * 2 + 0]);
        amat.y = sc * (hi ? (b_ih[n] + b_hh[n]) : W_hh[n * 2 + 1]);
    } else {
        amat.x = 0.0f; amat.y = 0.0f;     // zero gate rows -> hi-lane D == 0
    }

    const float wo0 = W_out[0], wo1 = W_out[1], bo = b_out[0];
    const v8f cz = {};                                 // zero C accumulator

    float h0 = 0.0f, h1 = 0.0f, c0 = 0.0f, c1 = 0.0f; // state, lane-major

    float* xb = xbuf[w];
    float* lb = lbuf[w];
    float* pb = pbuf[w];
    const int strow = lane * XP4;   // out-staging row (rows 16-31 = dead rows)
    const int xrow  = n * XP4;      // x broadcast row (mirrored per half-wave)

    float* __restrict__ outL = out;
    float* __restrict__ outP = out + (size_t)B_SZ * T_SZ;

    // ---- prime the x register double-buffer with chunk 0 (coalesced)
    float xr[16];
    #pragma unroll
    for (int r = 0; r < 16; ++r)
        xr[r] = x[(size_t)(b0 + r) * T_SZ + lane];

    for (int t0 = 0; t0 < T_SZ; t0 += TCH) {
        // ---- spill current chunk's x registers into LDS
        #pragma unroll
        for (int r = 0; r < 16; ++r)
            xb[r * XP4 + lane] = xr[r];

        // ---- prefetch next chunk's x; in flight behind ~2K cycles of compute
        if (t0 + TCH < T_SZ) {
            #pragma unroll
            for (int r = 0; r < 16; ++r)
                xr[r] = x[(size_t)(b0 + r) * T_SZ + (t0 + TCH) + lane];
        }

        // ---- 32 recurrent steps, quad-pipelined x reads
        float4 xq = *reinterpret_cast<const float4*>(&xb[xrow]);
        for (int tc = 0; tc < TCH; tc += 4) {
            // issue next group's quad now; its wait hides behind 4 steps
            float4 xn = *reinterpret_cast<const float4*>(&xb[xrow + tc + 4]);
            float4 lq, pq;
            STEP(xq.x, lq.x, pq.x);
            STEP(xq.y, lq.y, pq.y);
            STEP(xq.z, lq.z, pq.z);
            STEP(xq.w, lq.w, pq.w);
            *reinterpret_cast<float4*>(&lb[strow + tc]) = lq;
            *reinterpret_cast<float4*>(&pb[strow + tc]) = pq;
            xq = xn;
        }

        // ---- flush outputs (valid rows 0-15), fully coalesced
        #pragma unroll
        for (int r = 0; r < 16; ++r) {
            size_t o = (size_t)(b0 + r) * T_SZ + t0 + lane;
            outL[o] = lb[r * XP4 + lane];
            outP[o] = pb[r * XP4 + lane];
        }
    }
}

extern "C" void kernel_launch(void* const* d_in, const int* in_sizes, int n_in,
                              void* d_out, int out_size, void* d_ws, size_t ws_size,
                              hipStream_t stream) {
    const float* x     = (const float*)d_in[0];
    const float* W_ih  = (const float*)d_in[1];
    const float* W_hh  = (const float*)d_in[2];
    const float* b_ih  = (const float*)d_in[3];
    const float* b_hh  = (const float*)d_in[4];
    const float* W_out = (const float*)d_in[5];
    const float* b_out = (const float*)d_in[6];

    dim3 grid(B_SZ / (16 * WAVES));   // 64 blocks
    dim3 block(32 * WAVES);           // 4 waves/block, 1 tile/wave -> 256 waves

    hipLaunchKernelGGL(lstm_wmma_kernel, grid, block, 0, stream,
                       x, W_ih, W_hh, b_ih, b_hh, W_out, b_out, (float*)d_out);
}